// TopKBalancedNoisyGate_51908974739638
// MI455X (gfx1250) — compile-verified
//
#include <hip/hip_runtime.h>
#include <hip/hip_bf16.h>
#include <math.h>

// TopKBalancedNoisyGate for MI455X (gfx1250), fp32 WMMA path.
// T=8192 tokens, D=4096, E=16 experts, k=2 (+1 threshold).
//
// d_out layout (float*, 32769 elements):
//   [0        .. 16383] top_k_indices (8192 x 2), stored as float(index)
//   [16384    .. 32767] top_k_scores  (8192 x 2)
//   [32768]             gate_loss
//
// ws layout (float*): per-tile partials, 512 tiles * 32 floats = 64 KB
//   ws[tile*32 + e]      = load partial (sum over 16 rows)
//   ws[tile*32 + 16 + e] = importance partial

typedef __attribute__((ext_vector_type(2))) float v2f;
typedef __attribute__((ext_vector_type(8))) float v8f;

#define TOK  8192
#define DIM  4096
#define NEXP 16
#define NTILE (TOK / 16)

__device__ __forceinline__ float phi_f(float z) {
  return 0.5f * (1.0f + erff(z * 0.7071067811865476f));
}

__launch_bounds__(32)
__global__ void gate_main(const float* __restrict__ x,
                          const float* __restrict__ gw1,
                          const float* __restrict__ gw2,
                          const float* __restrict__ nw,
                          const float* __restrict__ noise,
                          float* __restrict__ out,
                          float* __restrict__ ws) {
  // Per-wave LDS scratch tiles (16x16 f32 each)
  __shared__ float T1[256];  // h / logits
  __shared__ float T2[256];  // logits_gate -> importance matrix
  __shared__ float T3[256];  // logits_noise
  __shared__ float T4[256];  // noise_control
  __shared__ float T5[256];  // load matrix

  const int tile = blockIdx.x;
  const int row0 = tile * 16;
  const int lane = threadIdx.x & 31;
  const int lr   = lane & 15;   // A: row-in-tile; B/C: expert column
  const int hi   = lane >> 4;   // half-wave selects K pair / row half
  const int koff = hi * 2;

  // ---- Fused dual GEMM over K = 4096 with V_WMMA_F32_16X16X4_F32 ----
  // A lane layout (16x4 f32): lanes 0-15 hold K={k,k+1}, lanes 16-31 K={k+2,k+3}
  // B lane layout (4x16 f32): lane n holds W[n][k..k+1] (same K split)
  const float* pA  = x   + (size_t)(row0 + lr) * DIM + koff;
  const float* pBg = gw1 + (size_t)lr * DIM + koff;
  const float* pBn = nw  + (size_t)lr * DIM + koff;

  v8f accg = {};  // x @ gate_w1.T  (C layout: lane=col, VGPR v = row v+8*hi)
  v8f accn = {};  // x @ noise_w.T

#pragma unroll 8
  for (int k = 0; k < DIM; k += 4) {
    v2f a  = *(const v2f*)(pA + k);
    v2f bg = *(const v2f*)(pBg + k);
    v2f bn = *(const v2f*)(pBn + k);
    accg = __builtin_amdgcn_wmma_f32_16x16x4_f32(false, a, false, bg,
                                                 (short)0, accg, false, false);
    accn = __builtin_amdgcn_wmma_f32_16x16x4_f32(false, a, false, bn,
                                                 (short)0, accn, false, false);
  }

  // ---- h = tanh(acc_g): C layout -> LDS (row-major 16x16) ----
#pragma unroll
  for (int v = 0; v < 8; ++v)
    T1[(v + 8 * hi) * 16 + lr] = tanhf(accg[v]);
  asm volatile("s_wait_dscnt 0x0" ::: "memory");

  // ---- GEMM2: logits_gate = h @ gate_w2.T (16x16x16 via 4 WMMAs) ----
  v8f lg = {};
  const float* pB2 = gw2 + lr * NEXP + koff;
#pragma unroll
  for (int kk = 0; kk < 4; ++kk) {
    v2f a2;
    a2.x = T1[lr * 16 + kk * 4 + koff];
    a2.y = T1[lr * 16 + kk * 4 + koff + 1];
    v2f b2 = *(const v2f*)(pB2 + kk * 4);
    lg = __builtin_amdgcn_wmma_f32_16x16x4_f32(false, a2, false, b2,
                                               (short)0, lg, false, false);
  }

  // ---- noise path + logits, write row-major tiles to LDS ----
#pragma unroll
  for (int v = 0; v < 8; ++v) {
    int m = v + 8 * hi;
    float z  = accn[v];
    float nc = (z > 20.0f ? z : log1pf(expf(z))) + 0.01f;  // softplus + eps
    float nz = noise[(size_t)(row0 + m) * NEXP + lr] * nc;
    float gv = lg[v];
    T1[m * 16 + lr] = gv + nz;  // logits
    T2[m * 16 + lr] = gv;       // logits_gate
    T3[m * 16 + lr] = nz;       // logits_noise
    T4[m * 16 + lr] = nc;       // noise_control
  }
  asm volatile("s_wait_dscnt 0x0" ::: "memory");

  // ---- per-row top-3, softmax(top-2), load matrix (lanes 0-15, row=lane) ----
  if (lane < 16) {
    const int r = lane;
    float v1 = -3.4e38f, v2 = -3.4e38f, v3 = -3.4e38f;
    int i1 = 0, i2 = 0;
#pragma unroll
    for (int e = 0; e < NEXP; ++e) {
      float val = T1[r * 16 + e];
      if (val > v1)      { v3 = v2; v2 = v1; i2 = i1; v1 = val; i1 = e; }
      else if (val > v2) { v3 = v2; v2 = val; i2 = e; }
      else if (val > v3) { v3 = val; }
    }
    // softmax over (v1, v2)
    float e2 = expf(v2 - v1);
    float den = 1.0f + e2;
    float s1 = 1.0f / den;
    float s2 = e2 / den;

    const int m = row0 + r;
    out[2 * m + 0] = (float)i1;
    out[2 * m + 1] = (float)i2;
    out[2 * TOK + 2 * m + 0] = s1;
    out[2 * TOK + 2 * m + 1] = s2;

    // load terms: thr_in = 3rd largest, thr_out = 2nd largest
#pragma unroll
    for (int e = 0; e < NEXP; ++e) {
      float gvv = T2[r * 16 + e];
      float ncv = T4[r * 16 + e];
      float nzv = T3[r * 16 + e];
      float p_in  = phi_f((gvv - v3) / ncv);
      float p_out = phi_f((gvv - v2) / ncv);
      T5[r * 16 + e] = (nzv > v3) ? p_in : p_out;
    }
    // importance matrix row (reuse T2 after all reads above)
#pragma unroll
    for (int e = 0; e < NEXP; ++e) T2[r * 16 + e] = 0.0f;
    T2[r * 16 + i1] = s1;
    T2[r * 16 + i2] = s2;
  }
  asm volatile("s_wait_dscnt 0x0" ::: "memory");

  // ---- fixed-order column sums -> deterministic per-tile partials ----
  if (lane < 16) {
    float sl = 0.0f, si = 0.0f;
#pragma unroll
    for (int r = 0; r < 16; ++r) {
      sl += T5[r * 16 + lane];
      si += T2[r * 16 + lane];
    }
    ws[tile * 32 + lane]      = sl;
    ws[tile * 32 + 16 + lane] = si;
  }
}

__launch_bounds__(32)
__global__ void gate_finalize(const float* __restrict__ ws,
                              float* __restrict__ out) {
  __shared__ float imp[NEXP];
  __shared__ float ld[NEXP];
  const int lane = threadIdx.x;
  if (lane < NEXP) {
    float sl = 0.0f, si = 0.0f;
    for (int t = 0; t < NTILE; ++t) {
      sl += ws[t * 32 + lane];
      si += ws[t * 32 + 16 + lane];
    }
    ld[lane]  = sl;
    imp[lane] = si;
  }
  asm volatile("s_wait_dscnt 0x0" ::: "memory");
  if (lane == 0) {
    float mi = 0.0f, ml = 0.0f;
    for (int e = 0; e < NEXP; ++e) { mi += imp[e]; ml += ld[e]; }
    mi *= (1.0f / NEXP);
    ml *= (1.0f / NEXP);
    float vi = 0.0f, vl = 0.0f;
    for (int e = 0; e < NEXP; ++e) {
      float di = imp[e] - mi; vi += di * di;
      float dl = ld[e]  - ml; vl += dl * dl;
    }
    vi *= (1.0f / (NEXP - 1));  // ddof=1
    vl *= (1.0f / (NEXP - 1));
    float cvi = vi / (mi * mi + 1e-10f);
    float cvl = vl / (ml * ml + 1e-10f);
    out[4 * TOK] = (cvi + cvl) * 0.01f;
  }
}

extern "C" void kernel_launch(void* const* d_in, const int* in_sizes, int n_in,
                              void* d_out, int out_size, void* d_ws, size_t ws_size,
                              hipStream_t stream) {
  const float* x   = (const float*)d_in[0];
  const float* gw1 = (const float*)d_in[1];
  const float* gw2 = (const float*)d_in[2];
  const float* nw  = (const float*)d_in[3];
  const float* nz  = (const float*)d_in[4];
  float* out = (float*)d_out;
  float* ws  = (float*)d_ws;

  gate_main<<<NTILE, 32, 0, stream>>>(x, gw1, gw2, nw, nz, out, ws);
  gate_finalize<<<1, 32, 0, stream>>>(ws, out);
}